// SimpleGCN_62843961475712
// MI455X (gfx1250) — compile-verified
//
#include <hip/hip_runtime.h>

typedef __attribute__((ext_vector_type(2))) float v2f;
typedef __attribute__((ext_vector_type(8))) float v8f;

#define N_NODES    100000
#define N_EDGES    1600000
#define IN_CH      128
#define HID_CH     64
#define OUT_CH     10
#define NUM_GRAPHS 256

// ---------------------------------------------------------------------------
// init: deg=1 (self loop), pool=0, cnt=0
// ---------------------------------------------------------------------------
__global__ void gcn_init(float* __restrict__ deg, float* __restrict__ pool,
                         float* __restrict__ cnt) {
  int i = blockIdx.x * blockDim.x + threadIdx.x;
  if (i < N_NODES) deg[i] = 1.0f;
  if (i < NUM_GRAPHS * HID_CH) pool[i] = 0.0f;
  if (i < NUM_GRAPHS) cnt[i] = 0.0f;
}

// ---------------------------------------------------------------------------
// in-degree accumulation over edges
// ---------------------------------------------------------------------------
__global__ void gcn_degree(const int* __restrict__ dst, float* __restrict__ deg) {
  int e = blockIdx.x * blockDim.x + threadIdx.x;
  if (e < N_EDGES) atomicAdd(&deg[dst[e]], 1.0f);
}

// deg -> rsqrt(deg) in place (deg >= 1 always, so no zero guard needed)
__global__ void gcn_dinv(float* __restrict__ deg) {
  int i = blockIdx.x * blockDim.x + threadIdx.x;
  if (i < N_NODES) deg[i] = rsqrtf(deg[i]);
}

// ---------------------------------------------------------------------------
// h = x @ W_conv via V_WMMA_F32_16X16X4_F32 (exact fp32 path).
// One wave computes a 16(M) x 64(N) output strip with 4 accumulators,
// looping K = 0..128 in steps of 4.
//
// A layout (16x4 f32): lanes 0-15: M=lane, v0=K+0, v1=K+1
//                      lanes 16-31: M=lane-16, v0=K+2, v1=K+3
// B layout (4x16 f32): lanes 0-15: N=lane, v0=row K+0, v1=row K+1
//                      lanes 16-31: N=lane-16, v0=row K+2, v1=row K+3
// D layout (16x16 f32): VGPR v: lanes 0-15 -> (M=v, N=lane),
//                               lanes 16-31 -> (M=v+8, N=lane-16)
// ---------------------------------------------------------------------------
__global__ __launch_bounds__(256) void gcn_gemm_wmma(
    const float* __restrict__ x,      // [N_NODES, IN_CH]
    const float* __restrict__ Wc,     // [IN_CH, HID_CH]
    float* __restrict__ h) {          // [N_NODES, HID_CH]
  const int wave = (blockIdx.x * blockDim.x + threadIdx.x) >> 5;
  const int lane = threadIdx.x & 31;
  const int m0 = wave * 16;
  if (m0 >= N_NODES) return;          // uniform per-wave; EXEC stays all-ones

  const int r  = lane & 15;           // M (for A) / N (for B/D) within tile
  const int hi = lane >> 4;           // 0: lanes 0-15, 1: lanes 16-31
  const float* __restrict__ xrow = x + (size_t)(m0 + r) * IN_CH;

  v8f acc0 = {}, acc1 = {}, acc2 = {}, acc3 = {};

  for (int k = 0; k < IN_CH; k += 4) {
    const int ka = k + (hi << 1);     // this lane-half's K pair base
    v2f a = *(const v2f*)(xrow + ka); // 8B-aligned (ka even, row 512B-aligned)

    const float* __restrict__ w0 = Wc + (size_t)ka * HID_CH;        // row ka
    const float* __restrict__ w1 = w0 + HID_CH;                     // row ka+1
    v2f b0, b1, b2, b3;
    b0.x = w0[r];      b0.y = w1[r];
    b1.x = w0[16 + r]; b1.y = w1[16 + r];
    b2.x = w0[32 + r]; b2.y = w1[32 + r];
    b3.x = w0[48 + r]; b3.y = w1[48 + r];

    acc0 = __builtin_amdgcn_wmma_f32_16x16x4_f32(false, a, false, b0, (short)0, acc0, false, false);
    acc1 = __builtin_amdgcn_wmma_f32_16x16x4_f32(false, a, false, b1, (short)0, acc1, false, false);
    acc2 = __builtin_amdgcn_wmma_f32_16x16x4_f32(false, a, false, b2, (short)0, acc2, false, false);
    acc3 = __builtin_amdgcn_wmma_f32_16x16x4_f32(false, a, false, b3, (short)0, acc3, false, false);
  }

  const int mout = m0 + (hi << 3);    // +8 for upper lane half
  float* __restrict__ hb = h + (size_t)mout * HID_CH + r;
#pragma unroll
  for (int v = 0; v < 8; ++v) {
    hb[(size_t)v * HID_CH +  0] = acc0[v];
    hb[(size_t)v * HID_CH + 16] = acc1[v];
    hb[(size_t)v * HID_CH + 32] = acc2[v];
    hb[(size_t)v * HID_CH + 48] = acc3[v];
  }
}

// ---------------------------------------------------------------------------
// agg = h * dinv^2 (self-loop message) + b_conv   -- replaces zero-init
// ---------------------------------------------------------------------------
__global__ void gcn_agg_init(const float* __restrict__ h,
                             const float* __restrict__ dinv,
                             const float* __restrict__ bconv,
                             float* __restrict__ agg) {
  int i = blockIdx.x * blockDim.x + threadIdx.x;
  if (i >= N_NODES * HID_CH) return;
  int n = i >> 6, c = i & 63;
  float di = dinv[n];
  agg[i] = h[i] * di * di + bconv[c];
}

// ---------------------------------------------------------------------------
// edge scatter: one thread per (edge, channel) -> coalesced h[src] reads,
// per-channel global_atomic_add_f32 into agg[dst]
// ---------------------------------------------------------------------------
__global__ void gcn_scatter(const int* __restrict__ src,
                            const int* __restrict__ dst,
                            const float* __restrict__ h,
                            const float* __restrict__ dinv,
                            float* __restrict__ agg) {
  unsigned int t = blockIdx.x * blockDim.x + threadIdx.x;
  unsigned int e = t >> 6;
  if (e >= N_EDGES) return;
  int c = t & 63;
  int s = src[e], d = dst[e];
  float w = dinv[s] * dinv[d];
  atomicAdd(&agg[(size_t)d * HID_CH + c], h[(size_t)s * HID_CH + c] * w);
}

// ---------------------------------------------------------------------------
// ReLU + segment-sum pool (batch is sorted, but atomics into 256*64 floats
// stay resident in L2 so contention is cheap)
// ---------------------------------------------------------------------------
__global__ void gcn_relu_pool(const float* __restrict__ agg,
                              const int* __restrict__ batch,
                              float* __restrict__ pool,
                              float* __restrict__ cnt) {
  int i = blockIdx.x * blockDim.x + threadIdx.x;
  if (i >= N_NODES * HID_CH) return;
  int n = i >> 6, c = i & 63;
  int g = batch[n];
  float v = fmaxf(agg[i], 0.0f);
  atomicAdd(&pool[g * HID_CH + c], v);
  if (c == 0) atomicAdd(&cnt[g], 1.0f);
}

// ---------------------------------------------------------------------------
// out[g,o] = (pool[g]/max(cnt,1)) @ W_lin + b_lin   (2560 outputs; trivial)
// ---------------------------------------------------------------------------
__global__ void gcn_final(const float* __restrict__ pool,
                          const float* __restrict__ cnt,
                          const float* __restrict__ Wl,
                          const float* __restrict__ bl,
                          float* __restrict__ out) {
  int i = blockIdx.x * blockDim.x + threadIdx.x;
  if (i >= NUM_GRAPHS * OUT_CH) return;
  int g = i / OUT_CH, o = i - g * OUT_CH;
  float inv = 1.0f / fmaxf(cnt[g], 1.0f);
  float s = bl[o];
#pragma unroll
  for (int hc = 0; hc < HID_CH; ++hc)
    s += pool[g * HID_CH + hc] * Wl[hc * OUT_CH + o];
  out[i] = s * 1.0f * inv + bl[o] * (1.0f - inv); // keep bias un-scaled
}

// NOTE on gcn_final: bias must not be divided by cnt. Rewritten cleanly below
// via a corrected kernel (this one is replaced in kernel_launch).
__global__ void gcn_final_fixed(const float* __restrict__ pool,
                                const float* __restrict__ cnt,
                                const float* __restrict__ Wl,
                                const float* __restrict__ bl,
                                float* __restrict__ out) {
  int i = blockIdx.x * blockDim.x + threadIdx.x;
  if (i >= NUM_GRAPHS * OUT_CH) return;
  int g = i / OUT_CH, o = i - g * OUT_CH;
  float inv = 1.0f / fmaxf(cnt[g], 1.0f);
  float s = 0.0f;
#pragma unroll
  for (int hc = 0; hc < HID_CH; ++hc)
    s += pool[g * HID_CH + hc] * Wl[hc * OUT_CH + o];
  out[i] = s * inv + bl[o];
}

extern "C" void kernel_launch(void* const* d_in, const int* in_sizes, int n_in,
                              void* d_out, int out_size, void* d_ws, size_t ws_size,
                              hipStream_t stream) {
  const float* x     = (const float*)d_in[0];          // [100000,128]
  const int*   eidx  = (const int*)d_in[1];            // [2,1600000]
  const int*   batch = (const int*)d_in[2];            // [100000]
  const float* Wc    = (const float*)d_in[3];          // [128,64]
  const float* bc    = (const float*)d_in[4];          // [64]
  const float* Wl    = (const float*)d_in[5];          // [64,10]
  const float* bl    = (const float*)d_in[6];          // [10]
  float* out = (float*)d_out;                          // [256,10]

  const int* src = eidx;
  const int* dst = eidx + N_EDGES;

  // workspace layout (floats)
  float* h    = (float*)d_ws;                          // 6,400,000
  float* agg  = h   + (size_t)N_NODES * HID_CH;        // 6,400,000
  float* deg  = agg + (size_t)N_NODES * HID_CH;        // 100,000 (-> dinv)
  float* pool = deg + N_NODES;                         // 16,384
  float* cnt  = pool + NUM_GRAPHS * HID_CH;            // 256

  const int B = 256;

  // 1) deg=1 / pool=0 / cnt=0
  gcn_init<<<(N_NODES + B - 1) / B, B, 0, stream>>>(deg, pool, cnt);
  // 2) in-degree over edges
  gcn_degree<<<(N_EDGES + B - 1) / B, B, 0, stream>>>(dst, deg);
  // 3) dinv = rsqrt(deg)
  gcn_dinv<<<(N_NODES + B - 1) / B, B, 0, stream>>>(deg);
  // 4) h = x @ W_conv  (WMMA fp32; 8 waves/block, 16 rows per wave)
  {
    int waves = N_NODES / 16;                          // 6250, exact
    int blocks = (waves + 7) / 8;
    gcn_gemm_wmma<<<blocks, B, 0, stream>>>(x, Wc, h);
  }
  // 5) agg = h*dinv^2 + b_conv
  gcn_agg_init<<<(N_NODES * HID_CH + B - 1) / B, B, 0, stream>>>(h, deg, bc, agg);
  // 6) edge scatter-add
  {
    long long work = (long long)N_EDGES * HID_CH;
    gcn_scatter<<<(unsigned)((work + B - 1) / B), B, 0, stream>>>(src, dst, h, deg, agg);
  }
  // 7) relu + pool
  gcn_relu_pool<<<(N_NODES * HID_CH + B - 1) / B, B, 0, stream>>>(agg, batch, pool, cnt);
  // 8) mean + final linear
  gcn_final_fixed<<<(NUM_GRAPHS * OUT_CH + B - 1) / B, B, 0, stream>>>(pool, cnt, Wl, bl, out);
}